// DynamicMemoryAllocation_60352880444049
// MI455X (gfx1250) — compile-verified
//
#include <hip/hip_runtime.h>

// Problem constants (match reference: B=4096, N=8192, R=4)
#define B_DIM 4096
#define N_DIM 8192
#define R_DIM 4
#define NTHREADS 512
#define EPT 16                  // elements per thread; wave owns 512 contiguous elems

// Branchless 64-bit compare-exchange. Keys are guaranteed distinct (unique
// index packed in low 32 bits), so (a>b)==asc is an exact swap predicate.
__device__ __forceinline__ void ce64(unsigned long long& a, unsigned long long& b, bool asc) {
  const bool sw = (a > b) == asc;
  const unsigned long long na = sw ? b : a;
  const unsigned long long nb = sw ? a : b;
  a = na;
  b = nb;
}

// Intra-wave compare-exchange at stride j = 16*JL via ds_swizzle xor-mask.
// offset encoding (group-of-32): [14:10]=xor, [9:5]=or, [4:0]=and.
template <int JL>
__device__ __forceinline__ void shuffle_ce(unsigned long long (&v)[EPT], int lane, bool asc) {
  const bool takeMin = (((lane & JL) == 0) == asc);  // lower-index lane gets min when asc
  #pragma unroll
  for (int e = 0; e < EPT; ++e) {
    const int lo = __builtin_amdgcn_ds_swizzle((int)(unsigned)(v[e] & 0xFFFFFFFFull),
                                               (JL << 10) | 0x1F);
    const int hi = __builtin_amdgcn_ds_swizzle((int)(unsigned)(v[e] >> 32),
                                               (JL << 10) | 0x1F);
    const unsigned long long other =
        ((unsigned long long)(unsigned)hi << 32) | (unsigned)lo;
    if ((v[e] > other) == takeMin) v[e] = other;
  }
}

// In-register bitonic-merge descent j = 8,4,2,1 with uniform direction.
__device__ __forceinline__ void reg_merge(unsigned long long (&v)[EPT], bool asc) {
  #pragma unroll
  for (int jj = 8; jj > 0; jj >>= 1) {
    #pragma unroll
    for (int e = 0; e < EPT; ++e) {
      const int m = e ^ jj;
      if (m > e) ce64(v[e], v[m], asc);
    }
  }
}

// Full intra-wave descent j = 256..16 (swizzles) then j = 8..1 (registers).
__device__ __forceinline__ void wave_tail_full(unsigned long long (&v)[EPT], int lane, bool asc) {
  shuffle_ce<16>(v, lane, asc);
  shuffle_ce<8>(v, lane, asc);
  shuffle_ce<4>(v, lane, asc);
  shuffle_ce<2>(v, lane, asc);
  shuffle_ce<1>(v, lane, asc);
  reg_merge(v, asc);
}

__global__ __launch_bounds__(NTHREADS)
void DynamicMemoryAllocation_kernel(const float* __restrict__ memory_usage,
                                    const float* __restrict__ free_gates,
                                    const float* __restrict__ write_weighting,
                                    const float* __restrict__ read_weightings,
                                    float* __restrict__ out) {
  // 64KB region: u64 keys during the sort, then reused as the allocation-weight
  // staging buffer for the coalesced write-out (keys are dead by then).
  __shared__ union SharedBuf {
    unsigned long long keys[N_DIM];
    float alloc[N_DIM];
  } s;
  __shared__ float s_part[NTHREADS];  // chunk-product scan buffer

  const int b    = blockIdx.x;
  const int tid  = threadIdx.x;
  const int lane = tid & 31;
  const int base = tid * EPT;         // 16-aligned contiguous chunk
  const size_t rowOff = (size_t)b * N_DIM;

  // Per-row free gates: block-uniform addresses -> scalar loads.
  const float fg[4] = {free_gates[b * R_DIM + 0], free_gates[b * R_DIM + 1],
                       free_gates[b * R_DIM + 2], free_gates[b * R_DIM + 3]};

  float* __restrict__ out_alloc = out;                          // (B,N) allocation_weights
  float* __restrict__ out_usage = out + (size_t)B_DIM * N_DIM;  // (B,N) new_usage

  unsigned long long v[EPT];

  // ---- Phase A: compute new_usage straight into registers (per-lane float4 loads;
  //      a wave covers a dense 2KB span per array, full cacheline utilization). ----
  {
    const float4* __restrict__ mu4 = reinterpret_cast<const float4*>(memory_usage + rowOff + base);
    const float4* __restrict__ ww4 = reinterpret_cast<const float4*>(write_weighting + rowOff + base);
    const float4* __restrict__ rw4 = reinterpret_cast<const float4*>(read_weightings) + (rowOff + base);
    float4* __restrict__ nu4 = reinterpret_cast<float4*>(out_usage + rowOff + base);
    #pragma unroll
    for (int g = 0; g < 4; ++g) {
      const float4 mu = mu4[g];
      const float4 ww = ww4[g];
      float4 nu;
      #pragma unroll
      for (int c = 0; c < 4; ++c) {
        const float4 rw = rw4[g * 4 + c];
        const float usage_read = (1.0f - rw.x * fg[0]) * (1.0f - rw.y * fg[1]) *
                                 (1.0f - rw.z * fg[2]) * (1.0f - rw.w * fg[3]);
        const float m = (&mu.x)[c];
        const float w = (&ww.x)[c];
        const float nuv = (m + w - m * w) * usage_read;  // in [0,1] -> monotone bits
        (&nu.x)[c] = nuv;
        v[g * 4 + c] = ((unsigned long long)__float_as_uint(nuv) << 32) |
                       (unsigned)(base + g * 4 + c);
      }
      nu4[g] = nu;
    }
  }

  // ---- Stages k=2..16: pure register sorting network ----
  #pragma unroll
  for (int kk = 2; kk <= 8; kk <<= 1) {
    #pragma unroll
    for (int jj = kk >> 1; jj > 0; jj >>= 1) {
      #pragma unroll
      for (int e = 0; e < EPT; ++e) {
        const int m = e ^ jj;
        if (m > e) ce64(v[e], v[m], ((e & kk) == 0));
      }
    }
  }
  reg_merge(v, ((tid & 1) == 0));  // k=16: direction (i&16)==0 <=> tid even

  // ---- Stages k=32..512: fully intra-wave (swizzles + registers), zero barriers ----
  {
    bool asc;
    asc = (((tid << 4) & 32) == 0);
    shuffle_ce<1>(v, lane, asc);
    reg_merge(v, asc);

    asc = (((tid << 4) & 64) == 0);
    shuffle_ce<2>(v, lane, asc);
    shuffle_ce<1>(v, lane, asc);
    reg_merge(v, asc);

    asc = (((tid << 4) & 128) == 0);
    shuffle_ce<4>(v, lane, asc);
    shuffle_ce<2>(v, lane, asc);
    shuffle_ce<1>(v, lane, asc);
    reg_merge(v, asc);

    asc = (((tid << 4) & 256) == 0);
    shuffle_ce<8>(v, lane, asc);
    shuffle_ce<4>(v, lane, asc);
    shuffle_ce<2>(v, lane, asc);
    shuffle_ce<1>(v, lane, asc);
    reg_merge(v, asc);

    asc = (((tid << 4) & 512) == 0);
    wave_tail_full(v, lane, asc);
  }

  // ---- Stages k=1024..8192: cross-wave steps (j>=512) as register-resident
  //      butterflies through LDS (elementwise partner = tid ^ (j>>4)); data
  //      never leaves registers otherwise. ----
  for (int k = 1024; k <= N_DIM; k <<= 1) {
    const bool asc = (((tid << 4) & k) == 0);
    for (int j = k >> 1; j >= 512; j >>= 1) {
      #pragma unroll
      for (int e = 0; e < EPT; ++e) s.keys[base + e] = v[e];  // own 128B chunk (b128 stores)
      __syncthreads();
      const int pbase = (tid ^ (j >> 4)) * EPT;               // partner's chunk
      const bool takeMin = ((((tid << 4) & j) == 0) == asc);  // lower half keeps min when asc
      #pragma unroll
      for (int e = 0; e < EPT; ++e) {
        const unsigned long long other = s.keys[pbase + e];
        if ((v[e] > other) == takeMin) v[e] = other;
      }
      __syncthreads();
    }
    wave_tail_full(v, lane, asc);
    // k == N_DIM: fully sorted chunk stays in registers for Phase C
  }

  // ---- Phase C: exclusive prefix product + allocation weights ----
  float chunk_prod = 1.0f;
  #pragma unroll
  for (int e = 0; e < EPT; ++e) {
    chunk_prod *= __uint_as_float((unsigned)(v[e] >> 32));
  }
  s_part[tid] = chunk_prod;
  __syncthreads();

  for (int off = 1; off < NTHREADS; off <<= 1) {  // Hillis-Steele product scan
    const float val = s_part[tid];
    const float pre = (tid >= off) ? s_part[tid - off] : 1.0f;
    __syncthreads();
    s_part[tid] = val * pre;
    __syncthreads();
  }
  float eprod = (tid == 0) ? 1.0f : s_part[tid - 1];  // exclusive prefix of this chunk

  // Scatter into LDS staging (keys are dead: all reads happened before the scan
  // barriers), then stream out coalesced b128 stores.
  #pragma unroll
  for (int e = 0; e < EPT; ++e) {
    const float key = __uint_as_float((unsigned)(v[e] >> 32));
    const unsigned idx = (unsigned)(v[e] & 0xFFFFFFFFu);
    s.alloc[idx] = (1.0f - key) * eprod;  // argsort is a permutation: each slot once
    eprod *= key;
  }
  __syncthreads();

  {
    const float4* __restrict__ sa4 = reinterpret_cast<const float4*>(s.alloc);
    float4* __restrict__ oa4 = reinterpret_cast<float4*>(out_alloc + rowOff);
    #pragma unroll
    for (int q = 0; q < N_DIM / 4 / NTHREADS; ++q) {  // 4 float4s per thread
      const int t = tid + q * NTHREADS;
      oa4[t] = sa4[t];
    }
  }
}

extern "C" void kernel_launch(void* const* d_in, const int* in_sizes, int n_in,
                              void* d_out, int out_size, void* d_ws, size_t ws_size,
                              hipStream_t stream) {
  (void)in_sizes; (void)n_in; (void)d_ws; (void)ws_size; (void)out_size;
  const float* memory_usage    = (const float*)d_in[0];  // (B,N)
  const float* free_gates      = (const float*)d_in[1];  // (B,R)
  const float* write_weighting = (const float*)d_in[2];  // (B,N)
  const float* read_weightings = (const float*)d_in[3];  // (B,N,R)
  float* out = (float*)d_out;                            // [alloc (B,N) | new_usage (B,N)]

  DynamicMemoryAllocation_kernel<<<dim3(B_DIM), dim3(NTHREADS), 0, stream>>>(
      memory_usage, free_gates, write_weighting, read_weightings, out);
}